// decoder_ds_35424890257710
// MI455X (gfx1250) — compile-verified
//
#include <hip/hip_runtime.h>
#include <math.h>

typedef __attribute__((ext_vector_type(2))) float v2f;
typedef __attribute__((ext_vector_type(8))) float v8f;

#define VOX 884736            // 96*96*96
#define N3  (3 * VOX)         // offset field elements
#define NB  (8 * VOX)         // batched voxels

// ---------------------------------------------------------------------------
// Kernel 1: alpha[b] = sigmoid( cw . (E_b @ fw) + c1_b * sum(fw) + f1_b )
// E viewed as [1024 x 125] (row r = b*128 + c). t[r] = sum_v E[r,v]*fw[v]
// computed with V_WMMA_F32_16X16X4_F32: 64 row tiles x 32 K-steps (K padded
// to 128 with zeros). fw is replicated across all 16 B-columns, so every
// column of the 16x16 D tile carries t for its row.
// K-steps 0..30 touch only K<=123 -> unguarded loads; the K=124..127 tail is
// peeled and built from one always-in-bounds load + a uniform select, so no
// divergent EXEC-masked loads appear in the WMMA loop.
// ---------------------------------------------------------------------------
__global__ __launch_bounds__(32)
void alpha_wmma_kernel(const float* __restrict__ E,
                       const float* __restrict__ cw,
                       const float* __restrict__ c1b,
                       const float* __restrict__ fw,
                       const float* __restrict__ f1b,
                       float* __restrict__ alpha_out)
{
    __shared__ float fw_s[128];
    __shared__ float t_s[1024];
    const int lane = threadIdx.x;              // single wave32

    for (int i = lane; i < 128; i += 32) fw_s[i] = (i < 125) ? fw[i] : 0.0f;
    __syncthreads();

    // A-matrix (16x4 f32) lane layout: M = lane%16; VGPR0 holds K = kb,
    // VGPR1 holds K = kb+1, where kb = 2*(lane>=16). B mirrors in K.
    const int M  = lane & 15;
    const int kb = (lane >> 4) << 1;

    for (int rt = 0; rt < 64; ++rt) {
        const float* Erow = E + (rt * 16 + M) * 125;
        v8f acc = {0.f, 0.f, 0.f, 0.f, 0.f, 0.f, 0.f, 0.f};
        #pragma unroll 1
        for (int kk = 0; kk < 31; ++kk) {      // K = kb + 4*kk, max 123 < 125
            const int k0 = (kk << 2) + kb;
            v2f a, b;
            a.x = Erow[k0];
            a.y = Erow[k0 + 1];
            b.x = fw_s[k0];
            b.y = fw_s[k0 + 1];
            acc = __builtin_amdgcn_wmma_f32_16x16x4_f32(
                false, a, false, b, (short)0, acc, false, false);
        }
        {   // tail: K = 124..127; only K = 124 (lower lane half, vgpr0) live
            const float e124 = Erow[124];      // always in bounds
            v2f a, b;
            a.x = (kb == 0) ? e124 : 0.0f;
            a.y = 0.0f;
            b.x = fw_s[124 + kb];              // kb=0 -> fw[124]; kb=2 -> 0
            b.y = fw_s[125 + kb];              // always 0 (padding)
            acc = __builtin_amdgcn_wmma_f32_16x16x4_f32(
                false, a, false, b, (short)0, acc, false, false);
        }
        // D layout: vgpr j, lanes 0-15 -> M=j ; lanes 16-31 -> M=j+8.
        // All N columns identical (fw replicated) -> read from N=0 lanes.
        if (M == 0) {
            const int mbase = rt * 16 + ((lane >> 4) << 3);
            #pragma unroll
            for (int j = 0; j < 8; ++j) t_s[mbase + j] = acc[j];
        }
    }
    __syncthreads();

    float sfw = 0.f;
    for (int i = lane; i < 125; i += 32) sfw += fw_s[i];
    #pragma unroll
    for (int off = 16; off > 0; off >>= 1) sfw += __shfl_xor(sfw, off, 32);

    const float cb = c1b[0];
    const float fb = f1b[0];
    for (int b = 0; b < 8; ++b) {
        float p = 0.f;
        for (int c = lane; c < 128; c += 32) p += cw[c] * t_s[b * 128 + c];
        #pragma unroll
        for (int off = 16; off > 0; off >>= 1) p += __shfl_xor(p, off, 32);
        if (lane == 0) {
            const float logit = p + cb * sfw + fb;
            alpha_out[b] = 1.0f / (1.0f + expf(-logit));   // ALPHA_M = 1
        }
    }
}

// ---------------------------------------------------------------------------
// Kernel 2: one separable pass of the 7-tap zero-padded mean filter along the
// axis selected by STRIDE (1 = X, 96 = Y, 9216 = Z). STRIDE is a template
// parameter so the coordinate extraction is a constant mul-shift, not a
// runtime integer division. Final pass also clamps to +-0.01 and
// pre-multiplies by the grid scale 96.
// ---------------------------------------------------------------------------
template <int STRIDE, bool FINAL>
__global__ __launch_bounds__(256)
void smooth_pass_kernel(const float* __restrict__ in, float* __restrict__ out)
{
    const int i = blockIdx.x * blockDim.x + threadIdx.x;
    if (i >= N3) return;
    const int c = (i / STRIDE) % 96;           // coordinate along filter axis
    float s = 0.0f;
    #pragma unroll
    for (int d = -3; d <= 3; ++d) {
        const int cc = c + d;
        if (cc >= 0 && cc < 96) s += in[i + d * STRIDE];
    }
    s *= (1.0f / 7.0f);
    if (FINAL) s = fminf(fmaxf(s, -0.01f), 0.01f) * 96.0f;
    out[i] = s;
}

// ---------------------------------------------------------------------------
// Kernel 3: coords = base + alpha[b] * field_scaled; trilinear-sample orig
// and mask with the reference's clamp/floor semantics. Outputs concatenated:
// def_ at [0, NB), mask_d at [NB, 2*NB).
// ---------------------------------------------------------------------------
__global__ __launch_bounds__(256)
void warp_sample_kernel(const float* __restrict__ orig,
                        const float* __restrict__ mask,
                        const float* __restrict__ field,
                        const float* __restrict__ alpha,
                        float* __restrict__ out)
{
    const int i = blockIdx.x * blockDim.x + threadIdx.x;
    if (i >= NB) return;
    const int b   = i / VOX;
    const int p   = i - b * VOX;
    const int z   = p / 9216;
    const int rem = p - z * 9216;
    const int y   = rem / 96;
    const int x   = rem - y * 96;

    const float a = alpha[b];
    float cz = (float)z + a * field[p];
    float cy = (float)y + a * field[VOX + p];
    float cx = (float)x + a * field[2 * VOX + p];
    cz = fminf(fmaxf(cz, 0.0f), 95.0f);
    cy = fminf(fmaxf(cy, 0.0f), 95.0f);
    cx = fminf(fmaxf(cx, 0.0f), 95.0f);

    const int z0 = (int)cz, y0 = (int)cy, x0 = (int)cx;   // coords >= 0
    const int z1 = min(z0 + 1, 95);
    const int y1 = min(y0 + 1, 95);
    const int x1 = min(x0 + 1, 95);
    const float fz = cz - (float)z0;
    const float fy = cy - (float)y0;
    const float fx = cx - (float)x0;
    const float gz = 1.0f - fz, gy = 1.0f - fy, gx = 1.0f - fx;

    const int r00 = z0 * 9216 + y0 * 96;
    const int r01 = z0 * 9216 + y1 * 96;
    const int r10 = z1 * 9216 + y0 * 96;
    const int r11 = z1 * 9216 + y1 * 96;

    const float* vb = orig + b * VOX;
    const float* mb = mask + b * VOX;

    float c00 = vb[r00 + x0] * gx + vb[r00 + x1] * fx;
    float c01 = vb[r01 + x0] * gx + vb[r01 + x1] * fx;
    float c10 = vb[r10 + x0] * gx + vb[r10 + x1] * fx;
    float c11 = vb[r11 + x0] * gx + vb[r11 + x1] * fx;
    out[i] = (c00 * gy + c01 * fy) * gz + (c10 * gy + c11 * fy) * fz;

    c00 = mb[r00 + x0] * gx + mb[r00 + x1] * fx;
    c01 = mb[r01 + x0] * gx + mb[r01 + x1] * fx;
    c10 = mb[r10 + x0] * gx + mb[r10 + x1] * fx;
    c11 = mb[r11 + x0] * gx + mb[r11 + x1] * fx;
    out[NB + i] = (c00 * gy + c01 * fy) * gz + (c10 * gy + c11 * fy) * fz;
}

// ---------------------------------------------------------------------------
extern "C" void kernel_launch(void* const* d_in, const int* in_sizes, int n_in,
                              void* d_out, int out_size, void* d_ws, size_t ws_size,
                              hipStream_t stream)
{
    const float* E    = (const float*)d_in[0];   // encode_ds [8,128,5,5,5]
    const float* orig = (const float*)d_in[1];   // [8,1,96,96,96]
    const float* mask = (const float*)d_in[2];   // [8,1,96,96,96]
    const float* cw   = (const float*)d_in[3];   // c1_w [128]
    const float* c1b  = (const float*)d_in[4];   // c1_b [1]
    const float* fw   = (const float*)d_in[5];   // f1_w [125]
    const float* f1b  = (const float*)d_in[6];   // f1_b [1]
    const float* off  = (const float*)d_in[7];   // offset_field [3,96,96,96]

    float* out   = (float*)d_out;
    float* field = (float*)d_ws;                 // N3 floats (final field)
    float* alpha = field + N3;                   // 8 floats
    float* scrA  = out;                          // d_out reused as scratch for
    float* scrB  = out + N3;                     // intermediate smooth passes

    alpha_wmma_kernel<<<1, 32, 0, stream>>>(E, cw, c1b, fw, f1b, alpha);

    const int bs = 256;
    const int g3 = (N3 + bs - 1) / bs;
    smooth_pass_kernel<9216, false><<<g3, bs, 0, stream>>>(off,  scrA);  // Z
    smooth_pass_kernel<96,   false><<<g3, bs, 0, stream>>>(scrA, scrB);  // Y
    smooth_pass_kernel<1,    true ><<<g3, bs, 0, stream>>>(scrB, field); // X + clamp*96

    const int gb = (NB + bs - 1) / bs;
    warp_sample_kernel<<<gb, bs, 0, stream>>>(orig, mask, field, alpha, out);
}